// EncoderSAGE_18262200942989
// MI455X (gfx1250) — compile-verified
//
#include <hip/hip_runtime.h>

typedef __attribute__((ext_vector_type(2))) float v2f;
typedef __attribute__((ext_vector_type(8))) float v8f;

#define IN_C 128
#define HID  64

// ---------------- zero fill ----------------
__global__ void sage_zero_kernel(float* __restrict__ p, long n) {
  long i = (long)blockIdx.x * blockDim.x + threadIdx.x;
  long stride = (long)gridDim.x * blockDim.x;
  for (; i < n; i += stride) p[i] = 0.0f;
}

// ---------------- in-degree (as float, for mean) ----------------
__global__ void sage_degree_kernel(const int* __restrict__ dst, int E,
                                   float* __restrict__ deg) {
  int i = blockIdx.x * blockDim.x + threadIdx.x;
  int stride = gridDim.x * blockDim.x;
  for (; i < E; i += stride) {
    atomicAdd(&deg[dst[i]], 1.0f);
  }
}

// ---------------- scatter-add of gathered features ----------------
// Each thread handles a 4-float chunk of one edge's feature vector.
template <int F, int CLOG2>  // F = feature width, 1<<CLOG2 = F/4 chunks
__global__ void sage_scatter_kernel(const float* __restrict__ x,
                                    const int* __restrict__ src,
                                    const int* __restrict__ dst, int E,
                                    float* __restrict__ agg) {
  long total = (long)E << CLOG2;
  long i = (long)blockIdx.x * blockDim.x + threadIdx.x;
  long stride = (long)gridDim.x * blockDim.x;
  for (; i < total; i += stride) {
    long e = i >> CLOG2;
    int f = (int)(i & ((1 << CLOG2) - 1)) << 2;
    int s = src[e];
    int d = dst[e];
    float4 v = *(const float4*)(x + (long)s * F + f);
    float* a = agg + (long)d * F + f;
    atomicAdd(a + 0, v.x);
    atomicAdd(a + 1, v.y);
    atomicAdd(a + 2, v.z);
    atomicAdd(a + 3, v.w);
  }
}

// ---------------- fused SAGE layer GEMM via f32 WMMA ----------------
// Out[i,:] = act( (Aagg[i,:]/max(deg[i],1)) @ Wl + bl + Aself[i,:] @ Wr )
// One wave computes a 16x16 tile of Out (rows = nodes, cols = HID slice).
//
// V_WMMA_F32_16X16X4_F32 operand layout (wave32):
//   A 16x4 : lane l (l&15 = M); VGPR r holds K = r + (l>=16 ? 2 : 0)
//   B 4x16 : lane l (l&15 = N); VGPR r holds K = r + (l>=16 ? 2 : 0)
//   C 16x16: lane l (l&15 = N); VGPR r holds M = r + (l>=16 ? 8 : 0)
template <int K, bool RELU>
__global__ void sage_wmma_gemm_kernel(const float* __restrict__ Aagg,
                                      const float* __restrict__ deg,
                                      const float* __restrict__ Aself,
                                      const float* __restrict__ Wl,
                                      const float* __restrict__ bl,
                                      const float* __restrict__ Wr,
                                      float* __restrict__ Out, int nNodes) {
  const int lane = threadIdx.x & 31;
  const int wid = (blockIdx.x * blockDim.x + threadIdx.x) >> 5;
  const int tilesN = HID / 16;  // 4 column tiles
  const int tile_m = wid / tilesN;
  const int tile_n = wid % tilesN;
  if (tile_m * 16 >= nNodes) return;  // wave-uniform: EXEC stays all-1s

  const int half = lane >> 4;   // 0: lanes 0-15, 1: lanes 16-31
  const int lm = lane & 15;
  const int koff = half * 2;    // K sub-offset for this lane half
  const int rowA = tile_m * 16 + lm;
  const int col = tile_n * 16 + lm;

  const float invc = 1.0f / fmaxf(deg[rowA], 1.0f);

  v8f c = {0.f, 0.f, 0.f, 0.f, 0.f, 0.f, 0.f, 0.f};

  const float* arow = Aagg + (long)rowA * K;
  const float* srow = Aself + (long)rowA * K;

  // neighbor-aggregate term: (agg/cnt) @ Wl
#pragma unroll
  for (int k = 0; k < K; k += 4) {
    v2f a;
    a.x = arow[k + koff] * invc;
    a.y = arow[k + koff + 1] * invc;
    v2f b;
    b.x = Wl[(k + koff) * HID + col];
    b.y = Wl[(k + koff + 1) * HID + col];
    c = __builtin_amdgcn_wmma_f32_16x16x4_f32(false, a, false, b, (short)0, c,
                                              false, false);
  }
  // self term: x @ Wr
#pragma unroll
  for (int k = 0; k < K; k += 4) {
    v2f a;
    a.x = srow[k + koff];
    a.y = srow[k + koff + 1];
    v2f b;
    b.x = Wr[(k + koff) * HID + col];
    b.y = Wr[(k + koff + 1) * HID + col];
    c = __builtin_amdgcn_wmma_f32_16x16x4_f32(false, a, false, b, (short)0, c,
                                              false, false);
  }

  const float bv = bl[col];
#pragma unroll
  for (int r = 0; r < 8; ++r) {
    float v = c[r] + bv;
    if (RELU) v = fmaxf(v, 0.0f);
    int row = tile_m * 16 + r + half * 8;
    Out[(long)row * HID + col] = v;
  }
}

extern "C" void kernel_launch(void* const* d_in, const int* in_sizes, int n_in,
                              void* d_out, int out_size, void* d_ws,
                              size_t ws_size, hipStream_t stream) {
  const float* x = (const float*)d_in[0];
  const int* ei = (const int*)d_in[1];
  const float* Wl1 = (const float*)d_in[2];
  const float* bl1 = (const float*)d_in[3];
  const float* Wr1 = (const float*)d_in[4];
  const float* Wl2 = (const float*)d_in[5];
  const float* bl2 = (const float*)d_in[6];
  const float* Wr2 = (const float*)d_in[7];
  float* out = (float*)d_out;

  const int N = in_sizes[0] / IN_C;  // 100000
  const int E = in_sizes[1] / 2;     // 1600000
  const int* src = ei;
  const int* dst = ei + E;

  // workspace layout (floats): deg[N] | agg1[N*IN_C] | h[N*HID] | agg2[N*HID]
  float* ws = (float*)d_ws;
  float* deg = ws;
  float* agg1 = deg + N;
  float* h = agg1 + (long)N * IN_C;
  float* agg2 = h + (long)N * HID;

  // zero deg + agg1 (contiguous) and agg2; h is fully overwritten by GEMM1
  sage_zero_kernel<<<2048, 256, 0, stream>>>(deg, (long)N * (1 + IN_C));
  sage_zero_kernel<<<2048, 256, 0, stream>>>(agg2, (long)N * HID);

  sage_degree_kernel<<<2048, 256, 0, stream>>>(dst, E, deg);

  // layer 1: scatter-mean of x (128 feats), then fused WMMA GEMM + ReLU
  sage_scatter_kernel<IN_C, 5><<<8192, 256, 0, stream>>>(x, src, dst, E, agg1);

  const int tilesM = (N + 15) / 16;
  const int totalWaves = tilesM * (HID / 16);
  const int wavesPerBlock = 256 / 32;
  const int gemmBlocks = (totalWaves + wavesPerBlock - 1) / wavesPerBlock;
  sage_wmma_gemm_kernel<IN_C, true><<<gemmBlocks, 256, 0, stream>>>(
      agg1, deg, x, Wl1, bl1, Wr1, h, N);

  // layer 2: scatter-mean of h (64 feats), then fused WMMA GEMM (no ReLU)
  sage_scatter_kernel<HID, 4><<<8192, 256, 0, stream>>>(h, src, dst, E, agg2);
  sage_wmma_gemm_kernel<HID, false><<<gemmBlocks, 256, 0, stream>>>(
      agg2, deg, h, Wl2, bl2, Wr2, out, N);
}